// ActorCriticPerSkill_53549652247151
// MI455X (gfx1250) — compile-verified
//
#include <hip/hip_runtime.h>

#define BATCH   32768
#define DOBS    64
#define NSKILL  16
#define HID     128
#define NACT    17
#define MAXTILES (BATCH / 16 + NSKILL)   // 2064
#define HP (HID + 4)                     // LDS row stride (floats), avoids bank conflicts

typedef __attribute__((ext_vector_type(2))) float v2f;
typedef __attribute__((ext_vector_type(8))) float v8f;

static __device__ __forceinline__ v8f wmma_f32_k4(v2f a, v2f b, v8f c) {
  // D(16x16,f32) = A(16x4,f32) * B(4x16,f32) + C
  return __builtin_amdgcn_wmma_f32_16x16x4_f32(
      /*neg_a=*/false, a, /*neg_b=*/false, b,
      /*c_mod=*/(short)0, c, /*reuse_a=*/false, /*reuse_b=*/false);
}

// ---------------- bucketing kernels ----------------

__global__ void ac_init_kernel(int* counts, int* cursor) {
  int i = threadIdx.x;
  if (i < NSKILL) { counts[i] = 0; cursor[i] = 0; }
}

__global__ void ac_count_kernel(const int* __restrict__ skill_ids, int* counts) {
  int i = blockIdx.x * blockDim.x + threadIdx.x;
  if (i < BATCH) atomicAdd(&counts[skill_ids[i]], 1);
}

__global__ void ac_scan_kernel(const int* __restrict__ counts, int* offsets,
                               int* totalTiles, int* tile_skill, int* tile_base,
                               int* tile_nrows) {
  // single thread: S = 16, trivial
  int off = 0, t = 0;
  for (int s = 0; s < NSKILL; ++s) {
    offsets[s] = off;
    int c = counts[s];
    for (int r = 0; r < c; r += 16) {
      tile_skill[t] = s;
      tile_base[t]  = off + r;
      int rem = c - r;
      tile_nrows[t] = rem < 16 ? rem : 16;
      ++t;
    }
    off += c;
  }
  offsets[NSKILL] = off;
  *totalTiles = t;
}

__global__ void ac_scatter_kernel(const int* __restrict__ skill_ids,
                                  const int* __restrict__ offsets,
                                  int* cursor, int* perm) {
  int i = blockIdx.x * blockDim.x + threadIdx.x;
  if (i < BATCH) {
    int s = skill_ids[i];
    int p = atomicAdd(&cursor[s], 1);
    perm[offsets[s] + p] = i;
  }
}

// ---------------- main WMMA kernel: one wave per 16-row tile ----------------

__global__ __launch_bounds__(32)
void ActorCriticPerSkill_wmma_kernel(
    const float* __restrict__ obs,
    const float* __restrict__ Wa1, const float* __restrict__ ba1,
    const float* __restrict__ Wa2, const float* __restrict__ ba2,
    const float* __restrict__ Wa3, const float* __restrict__ ba3,
    const float* __restrict__ Wc1, const float* __restrict__ bc1,
    const float* __restrict__ Wc2, const float* __restrict__ bc2,
    const float* __restrict__ Wc3, const float* __restrict__ bc3,
    const int* __restrict__ perm,
    const int* __restrict__ tile_skill,
    const int* __restrict__ tile_base,
    const int* __restrict__ tile_nrows,
    const int* __restrict__ totalTiles,
    float* __restrict__ out)   // [BATCH*NACT] logits, then [BATCH] values
{
  const int t = blockIdx.x;
  if (t >= *totalTiles) return;          // wave-uniform: EXEC stays all-ones

  const int s     = tile_skill[t];
  const int base  = tile_base[t];
  const int nrows = tile_nrows[t];
  const int lane  = threadIdx.x;
  const int nlo   = lane & 15;           // column / M index within half-wave
  const int khi   = (lane < 16) ? 0 : 2; // K sub-offset for A/B fragments
  const int mhi   = (lane < 16) ? 0 : 8; // M sub-offset for C/D rows

  __shared__ int   rows[16];
  __shared__ float hs1[16][HP];
  __shared__ float hs2[16][HP];

  if (lane < 16) rows[lane] = perm[base + ((lane < nrows) ? lane : 0)];
  __syncthreads();

  const v8f vzero = {0.f, 0.f, 0.f, 0.f, 0.f, 0.f, 0.f, 0.f};

  // ---- obs A-fragments (K = 0..63 in 16 steps of 4), reused by actor & critic L1.
  //      Fully static indexing (all k-loops fully unrolled) so the backend never
  //      falls back to v_movrels/m0 indexed VGPR access.
  const int arow = rows[nlo];
  v2f af[DOBS / 4];
  #pragma unroll
  for (int k = 0; k < DOBS / 4; ++k) {
    const float* p = obs + (size_t)arow * DOBS + 4 * k + khi;
    af[k].x = p[0];
    af[k].y = p[1];
  }

  v8f acc[8];

  // =================== ACTOR ===================
  // L1: [16x64] x [64x128]
  #pragma unroll
  for (int j = 0; j < 8; ++j) acc[j] = vzero;
  #pragma unroll
  for (int k = 0; k < DOBS / 4; ++k) {
    #pragma unroll
    for (int j = 0; j < 8; ++j) {
      const float* wp = Wa1 + ((size_t)s * DOBS + 4 * k + khi) * HID + 16 * j + nlo;
      v2f b; b.x = wp[0]; b.y = wp[HID];
      acc[j] = wmma_f32_k4(af[k], b, acc[j]);
    }
  }
  #pragma unroll
  for (int j = 0; j < 8; ++j) {
    const int n = 16 * j + nlo;
    const float bb = ba1[s * HID + n];
    #pragma unroll
    for (int r = 0; r < 8; ++r) {
      hs1[mhi + r][n] = tanhf(acc[j][r] + bb);
    }
  }
  __syncthreads();

  // L2: [16x128] x [128x128]
  #pragma unroll
  for (int j = 0; j < 8; ++j) acc[j] = vzero;
  #pragma unroll
  for (int k = 0; k < HID / 4; ++k) {
    v2f a; a.x = hs1[nlo][4 * k + khi]; a.y = hs1[nlo][4 * k + khi + 1];
    #pragma unroll
    for (int j = 0; j < 8; ++j) {
      const float* wp = Wa2 + ((size_t)s * HID + 4 * k + khi) * HID + 16 * j + nlo;
      v2f b; b.x = wp[0]; b.y = wp[HID];
      acc[j] = wmma_f32_k4(a, b, acc[j]);
    }
  }
  __syncthreads();
  #pragma unroll
  for (int j = 0; j < 8; ++j) {
    const int n = 16 * j + nlo;
    const float bb = ba2[s * HID + n];
    #pragma unroll
    for (int r = 0; r < 8; ++r) {
      hs2[mhi + r][n] = tanhf(acc[j][r] + bb);
    }
  }
  __syncthreads();

  // L3: [16x128] x [128x17] (pad N to 32 with zero columns)
  v8f acc3[2];
  acc3[0] = vzero; acc3[1] = vzero;
  #pragma unroll
  for (int k = 0; k < HID / 4; ++k) {
    v2f a; a.x = hs2[nlo][4 * k + khi]; a.y = hs2[nlo][4 * k + khi + 1];
    #pragma unroll
    for (int j = 0; j < 2; ++j) {
      const int n = 16 * j + nlo;
      const float* wp = Wa3 + ((size_t)s * HID + 4 * k + khi) * NACT + n;
      v2f b;
      b.x = (n < NACT) ? wp[0]    : 0.0f;
      b.y = (n < NACT) ? wp[NACT] : 0.0f;
      acc3[j] = wmma_f32_k4(a, b, acc3[j]);
    }
  }
  #pragma unroll
  for (int j = 0; j < 2; ++j) {
    const int n = 16 * j + nlo;
    if (n < NACT) {
      const float bb = ba3[s * NACT + n];
      #pragma unroll
      for (int r = 0; r < 8; ++r) {
        const int m = mhi + r;
        if (m < nrows) out[(size_t)rows[m] * NACT + n] = acc3[j][r] + bb;
      }
    }
  }
  __syncthreads();

  // =================== CRITIC ===================
  // L1: obs fragments reused
  #pragma unroll
  for (int j = 0; j < 8; ++j) acc[j] = vzero;
  #pragma unroll
  for (int k = 0; k < DOBS / 4; ++k) {
    #pragma unroll
    for (int j = 0; j < 8; ++j) {
      const float* wp = Wc1 + ((size_t)s * DOBS + 4 * k + khi) * HID + 16 * j + nlo;
      v2f b; b.x = wp[0]; b.y = wp[HID];
      acc[j] = wmma_f32_k4(af[k], b, acc[j]);
    }
  }
  #pragma unroll
  for (int j = 0; j < 8; ++j) {
    const int n = 16 * j + nlo;
    const float bb = bc1[s * HID + n];
    #pragma unroll
    for (int r = 0; r < 8; ++r) {
      hs1[mhi + r][n] = tanhf(acc[j][r] + bb);
    }
  }
  __syncthreads();

  // L2
  #pragma unroll
  for (int j = 0; j < 8; ++j) acc[j] = vzero;
  #pragma unroll
  for (int k = 0; k < HID / 4; ++k) {
    v2f a; a.x = hs1[nlo][4 * k + khi]; a.y = hs1[nlo][4 * k + khi + 1];
    #pragma unroll
    for (int j = 0; j < 8; ++j) {
      const float* wp = Wc2 + ((size_t)s * HID + 4 * k + khi) * HID + 16 * j + nlo;
      v2f b; b.x = wp[0]; b.y = wp[HID];
      acc[j] = wmma_f32_k4(a, b, acc[j]);
    }
  }
  __syncthreads();
  #pragma unroll
  for (int j = 0; j < 8; ++j) {
    const int n = 16 * j + nlo;
    const float bb = bc2[s * HID + n];
    #pragma unroll
    for (int r = 0; r < 8; ++r) {
      hs2[mhi + r][n] = tanhf(acc[j][r] + bb);
    }
  }
  __syncthreads();

  // Value head: [16x128] . [128x1] — plain dot per lane (N=1 too thin for WMMA)
  if (lane < nrows) {
    float sum = bc3[s];
    const float* w = Wc3 + (size_t)s * HID;
    #pragma unroll 8
    for (int k = 0; k < HID; ++k) sum += hs2[lane][k] * w[k];
    out[(size_t)BATCH * NACT + rows[lane]] = sum;
  }
}

// ---------------- launch ----------------

extern "C" void kernel_launch(void* const* d_in, const int* in_sizes, int n_in,
                              void* d_out, int out_size, void* d_ws, size_t ws_size,
                              hipStream_t stream) {
  const float* obs       = (const float*)d_in[0];
  const int*   skill_ids = (const int*)  d_in[1];
  const float* Wa1 = (const float*)d_in[2];
  const float* ba1 = (const float*)d_in[3];
  const float* Wa2 = (const float*)d_in[4];
  const float* ba2 = (const float*)d_in[5];
  const float* Wa3 = (const float*)d_in[6];
  const float* ba3 = (const float*)d_in[7];
  const float* Wc1 = (const float*)d_in[8];
  const float* bc1 = (const float*)d_in[9];
  const float* Wc2 = (const float*)d_in[10];
  const float* bc2 = (const float*)d_in[11];
  const float* Wc3 = (const float*)d_in[12];
  const float* bc3 = (const float*)d_in[13];
  float* out = (float*)d_out;

  int* ws         = (int*)d_ws;
  int* counts     = ws;                       // 16
  int* offsets    = ws + 16;                  // 17
  int* cursor     = ws + 33;                  // 16
  int* totalTiles = ws + 49;                  // 1
  int* tile_skill = ws + 50;                  // MAXTILES
  int* tile_base  = tile_skill + MAXTILES;    // MAXTILES
  int* tile_nrows = tile_base  + MAXTILES;    // MAXTILES
  int* perm       = tile_nrows + MAXTILES;    // BATCH

  ac_init_kernel<<<1, 32, 0, stream>>>(counts, cursor);
  ac_count_kernel<<<BATCH / 256, 256, 0, stream>>>(skill_ids, counts);
  ac_scan_kernel<<<1, 1, 0, stream>>>(counts, offsets, totalTiles,
                                      tile_skill, tile_base, tile_nrows);
  ac_scatter_kernel<<<BATCH / 256, 256, 0, stream>>>(skill_ids, offsets, cursor, perm);

  ActorCriticPerSkill_wmma_kernel<<<MAXTILES, 32, 0, stream>>>(
      obs, Wa1, ba1, Wa2, ba2, Wa3, ba3, Wc1, bc1, Wc2, bc2, Wc3, bc3,
      perm, tile_skill, tile_base, tile_nrows, totalTiles, out);
}